// DIN_84026740179500
// MI455X (gfx1250) — compile-verified
//
#include <hip/hip_runtime.h>
#include <hip/hip_bf16.h>
#include <math.h>

// DIN attention, MI455X (gfx1250, wave32).
// Folded layer-1:  din@w1 = q@(w1a+w1c) + f@(w1b-w1c) + (q*f)@w1d
// -> only the bilinear (q*f)@w1d term is per-(q,f); done with
//    V_WMMA_F32_16X16X4_F32 (fp32-exact).  Layer-2 (80->40) also WMMA.

typedef float v2f __attribute__((ext_vector_type(2)));
typedef float v8f __attribute__((ext_vector_type(8)));

#define B_  64
#define Q_  32
#define Fn_ 128
#define D_  128
#define H1_ 80
#define H2_ 40

__device__ __forceinline__ float sigmoidf(float x) {
  return 1.0f / (1.0f + __expf(-x));
}

// ---------------- Kernel 1: fold w1 / pack WMMA B fragments ----------------
// CmP layout: [nt(5)][kk(32)][lane(32)][v(2)], value = w1d[k][n],
//   k = 4*kk + 2*(lane>=16) + v,  n = 16*nt + (lane&15)   (B-matrix layout)
// W2P layout: [nt(3)][kk(20)][lane(32)][v(2)], N padded 40->48 with zeros.
__global__ __launch_bounds__(256) void fold_kernel(
    const float* __restrict__ w1, const float* __restrict__ w2,
    float* __restrict__ Af, float* __restrict__ Bf,
    float* __restrict__ CmP, float* __restrict__ W2P) {
  int idx = blockIdx.x * blockDim.x + threadIdx.x;
  if (idx < D_ * H1_) {
    int d = idx / H1_, n = idx % H1_;
    Af[idx] = w1[d * H1_ + n] + w1[(2 * D_ + d) * H1_ + n];
    Bf[idx] = w1[(D_ + d) * H1_ + n] - w1[(2 * D_ + d) * H1_ + n];
    int v = idx & 1, l = (idx >> 1) & 31, kk = (idx >> 6) & 31, nt = idx >> 11;
    int k  = 4 * kk + 2 * (l >> 4) + v;
    int n2 = 16 * nt + (l & 15);
    CmP[idx] = w1[(3 * D_ + k) * H1_ + n2];
  }
  if (idx < 3 * 20 * 64) {
    int v = idx & 1, l = (idx >> 1) & 31, rest = idx >> 6;
    int kk = rest % 20, nt = rest / 20;
    int k = 4 * kk + 2 * (l >> 4) + v;
    int n = 16 * nt + (l & 15);
    W2P[idx] = (n < H2_) ? w2[k * H2_ + n] : 0.0f;
  }
}

// ---------------- Kernel 2: QA[b,q,n] = q . Af  (2048 x 128 x 80) ----------
__global__ __launch_bounds__(256) void qa_kernel(
    const float* __restrict__ query, const float* __restrict__ Af,
    float* __restrict__ QA) {
  int idx = blockIdx.x * blockDim.x + threadIdx.x;
  if (idx >= B_ * Q_ * H1_) return;
  int row = idx / H1_, n = idx % H1_;
  const float* qp = query + (size_t)row * D_;
  float acc = 0.0f;
#pragma unroll 8
  for (int d = 0; d < D_; ++d) acc = fmaf(qp[d], Af[d * H1_ + n], acc);
  QA[idx] = acc;
}

// ---------------- Kernel 3: FB[b,f,n] = b1 + f . Bf  (8192 x 128 x 80) -----
__global__ __launch_bounds__(256) void fb_kernel(
    const float* __restrict__ facts, const float* __restrict__ Bf,
    const float* __restrict__ b1, float* __restrict__ FB) {
  int idx = blockIdx.x * blockDim.x + threadIdx.x;
  if (idx >= B_ * Fn_ * H1_) return;
  int row = idx / H1_, n = idx % H1_;
  const float* fp = facts + (size_t)row * D_;
  float acc = b1[n];
#pragma unroll 8
  for (int d = 0; d < D_; ++d) acc = fmaf(fp[d], Bf[d * H1_ + n], acc);
  FB[idx] = acc;
}

// ---------------- Kernel 4: main fused kernel, one block per (b,q) ---------
__global__ __launch_bounds__(256) void din_main_kernel(
    const float* __restrict__ query, const float* __restrict__ facts,
    const int* __restrict__ mask,
    const float* __restrict__ b2, const float* __restrict__ b3,
    const float* __restrict__ w3,
    const float* __restrict__ CmP, const float* __restrict__ W2P,
    const float* __restrict__ QA, const float* __restrict__ FB,
    float* __restrict__ out) {
  __shared__ __align__(16) float qs[D_];        // query row
  __shared__ float qa_s[H1_];                   // folded q@ term
  __shared__ float h1[Fn_ * 81];                // 128 x 80 (stride 81, cf-free)
  __shared__ float sc[Fn_];                     // scores / softmax probs
  __shared__ float red[Fn_];                    // reductions

  const int bq  = blockIdx.x;
  const int b   = bq >> 5;          // B=64
  const int tid = threadIdx.x;
  const int w    = tid >> 5;        // wave id, 8 waves, f-tile per wave
  const int lane = tid & 31;
  const int ln   = lane & 15;       // M/N index within tile
  const int khalf = ((lane >> 4) & 1) << 1;   // K offset: 0 or 2
  const int rbase = (lane >> 4) << 3;         // C-row offset: 0 or 8
  const int f0 = w << 4;

  if (tid < D_)  qs[tid]   = query[(size_t)bq * D_ + tid];
  if (tid < H1_) qa_s[tid] = QA[(size_t)bq * H1_ + tid];
  __syncthreads();

  // ---- layer 1: (q*f) @ w1d via WMMA f32 16x16x4, M=f-tile, N=80, K=128 ----
  v8f c[5];
#pragma unroll
  for (int i = 0; i < 5; ++i)
#pragma unroll
    for (int j = 0; j < 8; ++j) c[i][j] = 0.0f;

  const float* factrow = facts + ((size_t)(b * Fn_) + f0 + ln) * D_;
  for (int kk = 0; kk < 32; ++kk) {
    int d0 = 4 * kk + khalf;
    v2f a;
    a.x = qs[d0]     * factrow[d0];
    a.y = qs[d0 + 1] * factrow[d0 + 1];
#pragma unroll
    for (int nt = 0; nt < 5; ++nt) {
      v2f bb = *reinterpret_cast<const v2f*>(CmP + ((((nt << 5) + kk) << 5) + lane) * 2);
      c[nt] = __builtin_amdgcn_wmma_f32_16x16x4_f32(
          false, a, false, bb, (short)0, c[nt], false, false);
    }
  }
  // epilogue: + FB + QA, sigmoid -> h1 (wave-local rows f0..f0+15)
#pragma unroll
  for (int nt = 0; nt < 5; ++nt) {
    int n = (nt << 4) + ln;
    float qan = qa_s[n];
#pragma unroll
    for (int r = 0; r < 8; ++r) {
      int fl = f0 + rbase + r;
      float v = c[nt][r] + qan + FB[((size_t)(b * Fn_) + fl) * H1_ + n];
      h1[fl * 81 + n] = sigmoidf(v);
    }
  }

  // ---- layer 2: h1 @ w2 via WMMA, M=f-tile, N=48(pad of 40), K=80 ----------
  v8f e[3];
#pragma unroll
  for (int i = 0; i < 3; ++i)
#pragma unroll
    for (int j = 0; j < 8; ++j) e[i][j] = 0.0f;

  for (int kk = 0; kk < 20; ++kk) {
    int k0 = 4 * kk + khalf;
    v2f a;
    a.x = h1[(f0 + ln) * 81 + k0];
    a.y = h1[(f0 + ln) * 81 + k0 + 1];
#pragma unroll
    for (int nt = 0; nt < 3; ++nt) {
      v2f bb = *reinterpret_cast<const v2f*>(W2P + (((nt * 20 + kk) << 5) + lane) * 2);
      e[nt] = __builtin_amdgcn_wmma_f32_16x16x4_f32(
          false, a, false, bb, (short)0, e[nt], false, false);
    }
  }
  // + b2, sigmoid; h2 overwrites h1 columns 0..39 (same wave rows -> in order)
#pragma unroll
  for (int nt = 0; nt < 3; ++nt) {
    int n = (nt << 4) + ln;
    if (n < H2_) {
      float b2n = b2[n];
#pragma unroll
      for (int r = 0; r < 8; ++r) {
        int fl = f0 + rbase + r;
        h1[fl * 81 + n] = sigmoidf(e[nt][r] + b2n);
      }
    }
  }
  __syncthreads();

  // ---- layer 3 + mask: one thread per f row --------------------------------
  if (tid < Fn_) {
    float s3 = b3[0];
#pragma unroll 8
    for (int j = 0; j < H2_; ++j) s3 = fmaf(w3[j], h1[tid * 81 + j], s3);
    sc[tid]  = (mask[b * Fn_ + tid] == 1) ? s3 : -4294967295.0f;
    red[tid] = sc[tid];
  }
  __syncthreads();

  // ---- softmax over f ------------------------------------------------------
  for (int s = 64; s > 0; s >>= 1) {
    if (tid < s) red[tid] = fmaxf(red[tid], red[tid + s]);
    __syncthreads();
  }
  float mx = red[0];
  __syncthreads();
  if (tid < Fn_) {
    float ev = __expf(sc[tid] - mx);
    sc[tid]  = ev;
    red[tid] = ev;
  }
  __syncthreads();
  for (int s = 64; s > 0; s >>= 1) {
    if (tid < s) red[tid] += red[tid + s];
    __syncthreads();
  }
  float denom = red[0];
  __syncthreads();

  // ---- out[b,q] = sum_f p_f * (q . facts[b,f]) -----------------------------
  if (tid < Fn_) {
    const float4* fr = reinterpret_cast<const float4*>(
        facts + ((size_t)(b * Fn_) + tid) * D_);
    const float4* qv = reinterpret_cast<const float4*>(qs);
    float dq = 0.0f;
#pragma unroll 8
    for (int d4 = 0; d4 < D_ / 4; ++d4) {
      float4 fv = fr[d4], q4 = qv[d4];
      dq = fmaf(q4.x, fv.x, dq);
      dq = fmaf(q4.y, fv.y, dq);
      dq = fmaf(q4.z, fv.z, dq);
      dq = fmaf(q4.w, fv.w, dq);
    }
    red[tid] = (sc[tid] / denom) * dq;
  }
  __syncthreads();
  for (int s = 64; s > 0; s >>= 1) {
    if (tid < s) red[tid] += red[tid + s];
    __syncthreads();
  }
  if (tid == 0) out[bq] = red[0];
}

// ---------------------------------------------------------------------------
extern "C" void kernel_launch(void* const* d_in, const int* in_sizes, int n_in,
                              void* d_out, int out_size, void* d_ws, size_t ws_size,
                              hipStream_t stream) {
  const float* query = (const float*)d_in[0];
  const float* facts = (const float*)d_in[1];
  const int*   mask  = (const int*)d_in[2];
  const float* w1    = (const float*)d_in[3];
  const float* b1    = (const float*)d_in[4];
  const float* w2    = (const float*)d_in[5];
  const float* b2    = (const float*)d_in[6];
  const float* w3    = (const float*)d_in[7];
  const float* b3    = (const float*)d_in[8];
  float* out = (float*)d_out;

  float* ws  = (float*)d_ws;
  float* Af  = ws;                 // 128*80   = 10240
  float* Bf  = Af  + 10240;        // 10240
  float* CmP = Bf  + 10240;        // 10240 (packed WMMA B for layer 1)
  float* W2P = CmP + 10240;        // 3840  (packed WMMA B for layer 2)
  float* QA  = W2P + 3840;         // 2048*80  = 163840
  float* FB  = QA  + 163840;       // 8192*80  = 655360   (total ~3.4 MB)

  fold_kernel<<<40, 256, 0, stream>>>(w1, w2, Af, Bf, CmP, W2P);
  qa_kernel<<<(B_ * Q_ * H1_ + 255) / 256, 256, 0, stream>>>(query, Af, QA);
  fb_kernel<<<(B_ * Fn_ * H1_ + 255) / 256, 256, 0, stream>>>(facts, Bf, b1, FB);
  din_main_kernel<<<B_ * Q_, 256, 0, stream>>>(
      query, facts, mask, b2, b3, w3, CmP, W2P, QA, FB, out);
}